// Mapper_16638703305122
// MI455X (gfx1250) — compile-verified
//
#include <hip/hip_runtime.h>

// Problem constants (from reference)
#define DIM 1024
#define SEQ 2048
#define BZ 32
#define NUM_LS 8
#define DICT_LEN 9

// Tiling
#define BM 128
#define BN 128
#define BK 16
#define LDSW 20          // LDS row stride in floats: 80B (16B-aligned rows,
                         // conflict-free b64 fragment reads: l*20 mod 64 distinct)
#define NTHREADS 256     // 8 waves of 32

typedef __attribute__((ext_vector_type(2))) float v2f;
typedef __attribute__((ext_vector_type(4))) float v4f;
typedef __attribute__((ext_vector_type(8))) float v8f;
typedef int v4i __attribute__((vector_size(16)));  // matches builtin param type

#if defined(__has_builtin) && __has_builtin(__builtin_amdgcn_global_load_async_to_lds_b128)
#define HAVE_ASYNC_LDS 1
#else
#define HAVE_ASYNC_LDS 0
#endif
#if defined(__has_builtin) && __has_builtin(__builtin_amdgcn_s_wait_asynccnt)
#define HAVE_WAIT_ASYNC 1
#else
#define HAVE_WAIT_ASYNC 0
#endif

// Pointer casts to the exact pointee/address-space the async builtin wants:
// (v4i addrspace(1)*, v4i addrspace(3)*, imm offset, imm cpol)
#define GPTR(p) ((__attribute__((address_space(1))) v4i*)(p))
#define LPTR(p) ((__attribute__((address_space(3))) v4i*)(p))

__global__ __launch_bounds__(NTHREADS) void mapper_gemm_kernel(
    const float* __restrict__ x,        // [SEQ, BZ, DIM]
    const int*   __restrict__ lang_ids, // [BZ]
    const float* __restrict__ W,        // [NUM_LS, DIM, DIM] (torch: [out,in])
    const float* __restrict__ bias,     // [NUM_LS, DIM]
    float*       __restrict__ out)      // [SEQ, BZ, DIM]
{
    const int ntile = blockIdx.x;  // 0..7   (N direction)
    const int mtile = blockIdx.y;  // 0..15  (M direction, SEQ)
    const int bcol  = blockIdx.z;  // 0..31  (batch column)

    const int s0 = mtile * BM;
    const int n0 = ntile * BN;
    const int tid = threadIdx.x;

    const int gid    = DICT_LEN - 1 - lang_ids[bcol];
    const bool active = (gid >= 1) && (gid <= NUM_LS);
    int e = gid - 1;
    e = e < 0 ? 0 : (e > NUM_LS - 1 ? NUM_LS - 1 : e);

    if (!active) {
        // Pass-through: copy this 128x128 tile of x -> out with 128-bit ops.
        #pragma unroll
        for (int i = 0; i < 16; ++i) {
            int f   = i * NTHREADS + tid;   // 0..4095
            int row = f >> 5;               // 0..127
            int c4  = f & 31;               // 0..31 float4 per row
            size_t off = ((size_t)(s0 + row) * BZ + bcol) * DIM + n0 + c4 * 4;
            v4f v = *(const v4f*)(x + off);
            __builtin_nontemporal_store(v, (v4f*)(out + off));
        }
        return;
    }

    // ---- GEMM path: y[m, n] = sum_k x[m, k] * W[e][n, k] + bias[e][n] ----
    // Row-major LDS tiles (k contiguous) so (a) async b128 fills land directly
    // and (b) each WMMA fragment {K, K+1} is one aligned ds_load_b64.
    __shared__ float As[2][BM][LDSW];   // As[buf][m][k]
    __shared__ float Bs[2][BM][LDSW];   // Bs[buf][n][k] = W[e][n][k]

    const float* xq = x + (size_t)bcol * DIM;                         // + m*(BZ*DIM) + k
    const float* wq = W + (size_t)e * (DIM * DIM) + (size_t)n0 * DIM; // + n*DIM + k

    const int lane = tid & 31;
    const int hl   = lane >> 4;          // 0: K pair {0,1}, 1: K pair {2,3}
    const int l16  = lane & 15;
    const int wave = tid >> 5;           // 0..7
    const int wm   = (wave & 3) * 32;    // wave M offset within block tile
    const int wn   = (wave >> 2) * 64;   // wave N offset within block tile

    v8f acc[2][4] = {};                  // 2 M-subtiles x 4 N-subtiles of 16x16

    // Issue one stage's fills: 128 rows x 16 floats per matrix = 512 b128
    // transfers per matrix; 2 per thread per matrix.
    auto issue_stage = [&](int buf, int k0) {
        #pragma unroll
        for (int i = 0; i < 2; ++i) {
            int f   = i * NTHREADS + tid;  // 0..511
            int row = f >> 2;              // 0..127
            int kc  = f & 3;               // float4 slot along K
            const float* ga = xq + (size_t)(s0 + row) * (BZ * DIM) + k0 + kc * 4;
            const float* gb = wq + (size_t)row * DIM + k0 + kc * 4;
#if HAVE_ASYNC_LDS
            __builtin_amdgcn_global_load_async_to_lds_b128(
                GPTR(ga), LPTR(&As[buf][row][kc * 4]), 0, 0);
            __builtin_amdgcn_global_load_async_to_lds_b128(
                GPTR(gb), LPTR(&Bs[buf][row][kc * 4]), 0, 0);
#else
            v4f a = *(const v4f*)ga;
            v4f b = *(const v4f*)gb;
            *(v4f*)&As[buf][row][kc * 4] = a;
            *(v4f*)&Bs[buf][row][kc * 4] = b;
#endif
        }
    };
    auto wait_stage = [&]() {
#if HAVE_ASYNC_LDS
#if HAVE_WAIT_ASYNC
        __builtin_amdgcn_s_wait_asynccnt(0);
#else
        asm volatile("s_wait_asynccnt 0" ::: "memory");
#endif
#endif
    };

    issue_stage(0, 0);
    wait_stage();
    __syncthreads();

    const int NSTAGE = DIM / BK;  // 64
    for (int t = 0; t < NSTAGE; ++t) {
        const int cur   = t & 1;
        const bool more = (t + 1) < NSTAGE;

        // Start next stage's async fills into the other buffer (its readers
        // all retired before the barrier that ended iteration t-1).
        if (more) issue_stage(cur ^ 1, (t + 1) * BK);

        // Compute on current buffer: BK=16 -> 4 k-steps of 4, 8 WMMAs each.
        #pragma unroll
        for (int kk = 0; kk < BK; kk += 4) {
            const int ksel = kk + hl * 2;   // per ISA 16x4 f32 A/B lane layout
            v2f af[2];
            #pragma unroll
            for (int mt = 0; mt < 2; ++mt)
                af[mt] = *(const v2f*)&As[cur][wm + mt * 16 + l16][ksel];
            v2f bf[4];
            #pragma unroll
            for (int nt = 0; nt < 4; ++nt)
                bf[nt] = *(const v2f*)&Bs[cur][wn + nt * 16 + l16][ksel];
            #pragma unroll
            for (int mt = 0; mt < 2; ++mt)
                #pragma unroll
                for (int nt = 0; nt < 4; ++nt)
                    acc[mt][nt] = __builtin_amdgcn_wmma_f32_16x16x4_f32(
                        false, af[mt], false, bf[nt],
                        (short)0, acc[mt][nt], false, false);
        }

        if (more) wait_stage();   // wave's own async fills for t+1 complete
        __syncthreads();          // all waves' fills visible / reads retired
    }

    // Epilogue: add bias, scatter per documented 16x16 f32 C/D layout:
    // VGPR r holds row (r + 8*hl), lane%16 holds column. NT stores keep L2
    // free for W (fully L2-resident) and reused x rows.
    #pragma unroll
    for (int nt = 0; nt < 4; ++nt) {
        const int gn = n0 + wn + nt * 16 + l16;
        const float bv = bias[e * DIM + gn];
        #pragma unroll
        for (int mt = 0; mt < 2; ++mt) {
            #pragma unroll
            for (int r = 0; r < 8; ++r) {
                const int gm = s0 + wm + mt * 16 + hl * 8 + r;
                __builtin_nontemporal_store(acc[mt][nt][r] + bv,
                    out + ((size_t)gm * BZ + bcol) * DIM + gn);
            }
        }
    }
}

extern "C" void kernel_launch(void* const* d_in, const int* in_sizes, int n_in,
                              void* d_out, int out_size, void* d_ws, size_t ws_size,
                              hipStream_t stream) {
    const float* x        = (const float*)d_in[0];
    const int*   lang_ids = (const int*)d_in[1];
    const float* W        = (const float*)d_in[2];
    const float* bias     = (const float*)d_in[3];
    float* out = (float*)d_out;

    dim3 grid(DIM / BN, SEQ / BM, BZ);  // (8, 16, 32)
    mapper_gemm_kernel<<<grid, NTHREADS, 0, stream>>>(x, lang_ids, W, bias, out);
}